// Static_Graph_Convolution_75093208203291
// MI455X (gfx1250) — compile-verified
//
#include <hip/hip_runtime.h>

#define NDIM 4096
#define BETA 0.5f

typedef __attribute__((ext_vector_type(2))) float v2f;
typedef __attribute__((ext_vector_type(4))) float v4f;
typedef __attribute__((ext_vector_type(8))) float v8f;
typedef __attribute__((ext_vector_type(4))) unsigned int u32x4;
typedef __attribute__((ext_vector_type(8))) int i32x8;
typedef __attribute__((ext_vector_type(4))) int i32x4;

// ---------------------------------------------------------------------------
// Kernel 1: inv_d[i] = 1 / (sum_j A[i][j] + 1)
// ---------------------------------------------------------------------------
__global__ __launch_bounds__(256)
void degree_kernel(const float* __restrict__ A, float* __restrict__ inv_d) {
    const int row = blockIdx.x;
    const float* a = A + (size_t)row * NDIM;

    float s = 0.f;
    for (int j = threadIdx.x * 4; j < NDIM; j += 256 * 4) {
        v4f v = *(const v4f*)(a + j);
        s += v.x + v.y + v.z + v.w;
    }
    for (int off = 16; off > 0; off >>= 1)
        s += __shfl_down(s, off, 32);

    __shared__ float wsum[8];
    const int wave = threadIdx.x >> 5;
    const int lane = threadIdx.x & 31;
    if (lane == 0) wsum[wave] = s;
    __syncthreads();
    if (threadIdx.x == 0) {
        float t = 0.f;
        for (int w = 0; w < 8; ++w) t += wsum[w];
        inv_d[row] = 1.0f / (t + 1.0f);
    }
}

// ---------------------------------------------------------------------------
// Kernel 2: out = T @ W,  T = BETA*h + (1-BETA)*(A+I)*diag(inv_d)
// 128x128 block tile, KT=32, 8 wave32 in 4x2 grid, 2x4 16x16 f32 C-tiles/wave.
// W tiles move via the Tensor Data Mover into a DOUBLE-BUFFERED LDS slot
// (DMA for stage s+1 runs under stage s's 64 WMMAs; wait tensorcnt<=1).
// T tile fused on the fly through VGPRs with a register software pipeline.
// ---------------------------------------------------------------------------
#define BM 128
#define BN 128
#define KT 32
#define TS_STRIDE 36   // [128][36]: float4-aligned rows, conflict-free A-frag reads
#define WS_STRIDE 132  // [32][132]: TDM pad_enable gives the +4 DWORD row pad
#define WS_ELEMS (KT * WS_STRIDE)

__device__ __forceinline__ void tdm_load_W(const float* __restrict__ Wg,
                                           float* WsBase, int k0, int n0) {
    // ---- Tensor DMA descriptor (D#), 2D tile: 32 rows x 128 cols of f32 ----
    unsigned long long gaddr =
        (unsigned long long)(const void*)(Wg + (size_t)k0 * NDIM + n0);
    unsigned lds = (unsigned)(unsigned long long)(void*)WsBase;

    u32x4 g0;
    g0.x = 1u;                                    // count=1, user mode, no gather
    g0.y = lds;                                   // lds_addr (bytes)
    g0.z = (unsigned)gaddr;                       // global_addr[31:0]
    g0.w = (unsigned)(gaddr >> 32) | (2u << 30);  // global_addr[56:32] | type=2

    i32x8 g1;
    g1[0] = (2 << 16)        // data_size = 4 bytes
          | (1 << 20)        // pad_enable
          | (6 << 22)        // pad_interval: 128 DWORDs between pads
          | (3 << 25);       // pad_amount: 4 DWORDs -> row stride 132
    g1[1] = (NDIM & 0xFFFF) << 16;                    // tensor_dim0[15:0]
    g1[2] = ((NDIM >> 16) & 0xFFFF)                   // tensor_dim0[31:16]
          | ((NDIM & 0xFFFF) << 16);                  // tensor_dim1[15:0]
    g1[3] = ((NDIM >> 16) & 0xFFFF) | (BN << 16);     // tensor_dim1 hi | tile_dim0=128
    g1[4] = KT;                                       // tile_dim1=32, tile_dim2=0
    g1[5] = NDIM;                                     // tensor_dim0_stride[31:0]
    g1[6] = 0;                                        // stride hi | dim1_stride lo
    g1[7] = 0;
    i32x4 gz4 = {0, 0, 0, 0};                         // groups 2/3 unused (2D tile)
    i32x8 gz8 = {0, 0, 0, 0, 0, 0, 0, 0};
    __builtin_amdgcn_tensor_load_to_lds(g0, g1, gz4, gz4, gz8, 0);
}

__global__ __launch_bounds__(256)
void fused_gemm_kernel(const float* __restrict__ A,
                       const float* __restrict__ H,
                       const float* __restrict__ W,
                       const float* __restrict__ inv_d,
                       float* __restrict__ out) {
    __shared__ float Ts[BM * TS_STRIDE];
    __shared__ float Ws[2 * WS_ELEMS];     // double-buffered TDM destination

    const int tid  = threadIdx.x;
    const int lane = tid & 31;
    const int wave = tid >> 5;
    const int wr   = wave & 3;            // wave row (32 output rows)
    const int wc   = wave >> 2;           // wave col (64 output cols)

    const int m0 = blockIdx.y * BM;
    const int n0 = blockIdx.x * BN;

    const int l16 = lane & 15;
    const int h2  = (lane >> 4) << 1;     // 0 or 2: K-pair selector

    // T-tile staging geometry (per thread: 4 float4 quads)
    int qm[4], qk[4];
    #pragma unroll
    for (int i = 0; i < 4; ++i) {
        const int idx = tid + i * 256;    // float4 index, 1024 total
        qm[i] = idx >> 3;                 // 8 float4 per row
        qk[i] = (idx & 7) << 2;           // col offset inside tile
    }

    v8f zero = {};
    v8f acc[2][4];
    for (int tm = 0; tm < 2; ++tm)
        for (int tn = 0; tn < 4; ++tn)
            acc[tm][tn] = zero;

    // ---- software-pipeline registers for the T tile ----
    v4f av[4], hv[4], dv[4];
    #pragma unroll
    for (int i = 0; i < 4; ++i) {
        const size_t goff = (size_t)(m0 + qm[i]) * NDIM + qk[i];
        av[i] = *(const v4f*)(A + goff);
        hv[i] = *(const v4f*)(H + goff);
        dv[i] = *(const v4f*)(inv_d + qk[i]);
    }

    // ---- prologue: kick off DMA for the first W tile ----
    if (wave == 0) tdm_load_W(W, Ws, 0, n0);

    for (int k0 = 0; k0 < NDIM; k0 += KT) {
        const int cur  = (k0 / KT) & 1;
        const bool more = (k0 + KT) < NDIM;

        __syncthreads();                   // stage k0-KT fully consumed

        // DMA for the NEXT W tile runs underneath this stage's WMMAs
        if (wave == 0 && more) tdm_load_W(W, Ws + (cur ^ 1) * WS_ELEMS, k0 + KT, n0);

        // ---- fused T-tile store: T = BETA*h + (1-BETA)*(A+I)*inv_d[col] ----
        #pragma unroll
        for (int i = 0; i < 4; ++i) {
            const int gcol = k0 + qk[i];
            const int grow = m0 + qm[i];
            v4f t;
            t.x = BETA * hv[i].x + (1.f - BETA) * (av[i].x + (grow == gcol + 0 ? 1.f : 0.f)) * dv[i].x;
            t.y = BETA * hv[i].y + (1.f - BETA) * (av[i].y + (grow == gcol + 1 ? 1.f : 0.f)) * dv[i].y;
            t.z = BETA * hv[i].z + (1.f - BETA) * (av[i].z + (grow == gcol + 2 ? 1.f : 0.f)) * dv[i].z;
            t.w = BETA * hv[i].w + (1.f - BETA) * (av[i].w + (grow == gcol + 3 ? 1.f : 0.f)) * dv[i].w;
            *(v4f*)(Ts + qm[i] * TS_STRIDE + qk[i]) = t;
        }

        // ---- prefetch next stage's A/h/inv_d quads (drain under WMMAs) ----
        if (more) {
            #pragma unroll
            for (int i = 0; i < 4; ++i) {
                const int kn = k0 + KT + qk[i];
                const size_t goff = (size_t)(m0 + qm[i]) * NDIM + kn;
                av[i] = *(const v4f*)(A + goff);
                hv[i] = *(const v4f*)(H + goff);
                dv[i] = *(const v4f*)(inv_d + kn);
            }
        }

        // only the OLDER DMA (current buffer) must have completed
        if (more) __builtin_amdgcn_s_wait_tensorcnt(1);
        else      __builtin_amdgcn_s_wait_tensorcnt(0);
        __syncthreads();                   // Ts stores + Ws[cur] DMA visible

        // ---- compute: 8 K-substeps of 4, 8 WMMA each ----
        const float* Wcur = Ws + cur * WS_ELEMS;
        #pragma unroll
        for (int kk = 0; kk < KT; kk += 4) {
            v2f a[2];
            v2f b[4];
            #pragma unroll
            for (int tm = 0; tm < 2; ++tm) {
                const float* p = Ts + (wr * 32 + tm * 16 + l16) * TS_STRIDE + kk + h2;
                a[tm].x = p[0];
                a[tm].y = p[1];
            }
            #pragma unroll
            for (int tn = 0; tn < 4; ++tn) {
                const float* p = Wcur + (kk + h2) * WS_STRIDE + wc * 64 + tn * 16 + l16;
                b[tn].x = p[0];
                b[tn].y = p[WS_STRIDE];
            }
            #pragma unroll
            for (int tm = 0; tm < 2; ++tm)
                #pragma unroll
                for (int tn = 0; tn < 4; ++tn)
                    acc[tm][tn] = __builtin_amdgcn_wmma_f32_16x16x4_f32(
                        false, a[tm], false, b[tn],
                        (short)0, acc[tm][tn], false, false);
        }
    }

    // ---- epilogue: C/D layout -> global, non-temporal (write-once data) ----
    const int rbase = m0 + wr * 32;
    const int cbase = n0 + wc * 64;
    const int rhalf = (lane >> 4) << 3;   // +8 rows for lanes 16..31
    #pragma unroll
    for (int tm = 0; tm < 2; ++tm)
        #pragma unroll
        for (int tn = 0; tn < 4; ++tn)
            #pragma unroll
            for (int r = 0; r < 8; ++r) {
                const int row = rbase + tm * 16 + r + rhalf;
                const int col = cbase + tn * 16 + l16;
                __builtin_nontemporal_store(acc[tm][tn][r],
                                            out + (size_t)row * NDIM + col);
            }
}

// ---------------------------------------------------------------------------
extern "C" void kernel_launch(void* const* d_in, const int* in_sizes, int n_in,
                              void* d_out, int out_size, void* d_ws, size_t ws_size,
                              hipStream_t stream) {
    const float* A = (const float*)d_in[0];
    const float* H = (const float*)d_in[1];
    const float* W = (const float*)d_in[2];
    float* out   = (float*)d_out;
    float* inv_d = (float*)d_ws;   // 4096 floats of scratch

    degree_kernel<<<NDIM, 256, 0, stream>>>(A, inv_d);

    dim3 grid(NDIM / BN, NDIM / BM);
    fused_gemm_kernel<<<grid, 256, 0, stream>>>(A, H, W, inv_d, out);
}